// Embedding_30013231464674
// MI455X (gfx1250) — compile-verified
//
#include <hip/hip_runtime.h>

// Embedding gather: out[t, :] = weight[token_ids[t], :]
// token_ids: int32[32768]  weight: float32[50257, 1024]  out: float32[32768, 1024]
//
// Pure memory-bound op (0 FLOPs). MI455X strategy:
//  - 128-bit loads/stores per lane (global_load_b128 / global_store_b128)
//  - weight loads: default temporal hint -> table (~196 MiB) stays hot in 192 MB L2
//  - output stores: NON-temporal (write-once stream must not evict the table)
//  - token ids: uniform per block -> scalar s_load clause
//  - 8 tokens per 256-thread block, phases split so 8 b128 loads are in
//    flight per thread before the first s_wait_loadcnt (MLP, not serialized)

typedef __attribute__((ext_vector_type(4))) float v4f;

#define D_MODEL 1024
#define THREADS 256            // 8 waves (wave32); 256 lanes * 16B == one 4KB row
#define TOKENS_PER_BLOCK 8

__global__ __launch_bounds__(THREADS)
void embed_gather_kernel(const int* __restrict__ token_ids,
                         const float* __restrict__ weight,
                         float* __restrict__ out,
                         int n_tokens)
{
    const int tid = threadIdx.x;                                   // 0..255
    const long base_tok = (long)blockIdx.x * TOKENS_PER_BLOCK;

    if (base_tok + TOKENS_PER_BLOCK <= (long)n_tokens) {
        // ---- fast path (always taken for 32768 tokens): no per-token branches ----

        // Phase 1: all token ids (uniform -> scalar load clause)
        int rows[TOKENS_PER_BLOCK];
#pragma unroll
        for (int t = 0; t < TOKENS_PER_BLOCK; ++t)
            rows[t] = token_ids[base_tok + t];

        // Phase 2: issue all 8 gathers back-to-back -> 8 outstanding b128 loads
        v4f v[TOKENS_PER_BLOCK];
#pragma unroll
        for (int t = 0; t < TOKENS_PER_BLOCK; ++t) {
            const v4f* __restrict__ src =
                (const v4f*)(weight + (long)rows[t] * D_MODEL) + tid;
            v[t] = *src;
        }

        // Phase 3: streaming NT stores (scheduler can count loadcnt down 7..0)
#pragma unroll
        for (int t = 0; t < TOKENS_PER_BLOCK; ++t) {
            v4f* __restrict__ dst = (v4f*)(out + (base_tok + t) * D_MODEL) + tid;
            __builtin_nontemporal_store(v[t], dst);
        }
    } else {
        // ---- guarded tail (never taken when n_tokens % TOKENS_PER_BLOCK == 0) ----
        for (int t = 0; t < TOKENS_PER_BLOCK; ++t) {
            const long tok = base_tok + t;
            if (tok >= (long)n_tokens) break;
            const int row = token_ids[tok];
            const v4f* __restrict__ src =
                (const v4f*)(weight + (long)row * D_MODEL) + tid;
            v4f val = *src;
            v4f* __restrict__ dst = (v4f*)(out + tok * D_MODEL) + tid;
            __builtin_nontemporal_store(val, dst);
        }
    }
}

extern "C" void kernel_launch(void* const* d_in, const int* in_sizes, int n_in,
                              void* d_out, int out_size, void* d_ws, size_t ws_size,
                              hipStream_t stream) {
    const int*   token_ids = (const int*)d_in[0];   // [8*4096] int32
    const float* weight    = (const float*)d_in[1]; // [50257*1024] f32
    float*       out       = (float*)d_out;         // [32768*1024] f32

    const int n_tokens = in_sizes[0];               // 32768
    const int grid = (n_tokens + TOKENS_PER_BLOCK - 1) / TOKENS_PER_BLOCK; // 4096

    embed_gather_kernel<<<grid, THREADS, 0, stream>>>(token_ids, weight, out, n_tokens);
}